// PINN_Burgers2D_30863634989174
// MI455X (gfx1250) — compile-verified
//
#include <hip/hip_runtime.h>
#include <hip/hip_bf16.h>

// CDNA5 (gfx1250) wave32 WMMA types
typedef float v2f __attribute__((ext_vector_type(2)));
typedef float v8f __attribute__((ext_vector_type(8)));

#define ROWS 16          // collocation points per workgroup (M tile)
#define COLS 256         // hidden width
#define RSTRIDE 264      // padded LDS row stride (floats) to spread banks
#define NCH 6            // forward-AD channels: val, dx, dy, dt, dxx, dyy
#define NTHREADS 256     // 8 waves -> 2 waves per SIMD32
#define HBUF (NCH * ROWS * RSTRIDE)          // floats per ping-pong buffer
#define MISC_IN4   0                         // 16 rows x 4 normalized inputs
#define MISC_NU    64                        // 16 raw nu values
#define MISC_PART  80                        // 16 rows x 16 slices x 12 partials
#define MISC_FLOATS (80 + 16 * 16 * 12)

// swish value + first/second derivative chain rule for one neuron
__device__ __forceinline__ void swish_d2(float z, float zx, float zy, float zt,
                                         float zxx, float zyy,
                                         float& h, float& hx, float& hy, float& ht,
                                         float& hxx, float& hyy) {
    float s   = 1.0f / (1.0f + __expf(-z));   // sigma(z)
    float sp  = s * (1.0f - s);               // sigma'
    float fp  = s + z * sp;                   // f'  = d(z*sigma)/dz
    float spp = sp * (1.0f - 2.0f * s);       // sigma''
    float fpp = 2.0f * sp + z * spp;          // f''
    h   = z * s;
    hx  = fp * zx;
    hy  = fp * zy;
    ht  = fp * zt;
    hxx = fpp * zx * zx + fp * zxx;
    hyy = fpp * zy * zy + fp * zyy;
}

// One 256->256 hidden layer for all 6 AD channels.
// W/B come straight from kernel args so weight loads stay on the GLOBAL path
// (LOADcnt only), decoupled from LDS DScnt waits. Wnext prefetched into GL2.
__device__ __forceinline__ void hidden_layer(
        const float* __restrict__ cur, float* __restrict__ nxt,
        const float* __restrict__ W, const float* __restrict__ B,
        const float* __restrict__ Wnext)
{
    const int tid   = threadIdx.x;
    const int lane  = tid & 31;
    const int wave  = tid >> 5;          // 8 waves
    const int mrow  = lane & 15;         // A-matrix M row for this lane
    const int koff  = (lane >> 4) * 2;   // lanes 16-31 hold K+2, K+3
    const int mbase = (lane >> 4) * 8;   // C/D: lanes 16-31 hold M = r+8

    // Speculative prefetch of the next layer's 256KB weight stream
    // (global_prefetch_b8; one 128B line per iteration, 2048 lines total).
    if (Wnext != nullptr) {
        #pragma unroll
        for (int p = 0; p < 8; ++p)
            __builtin_prefetch(&Wnext[(tid + p * NTHREADS) * 32], 0, 1);
    }

    #pragma unroll
    for (int i = 0; i < 2; ++i) {
        const int nt   = wave + 8 * i;           // column tile 0..15
        const int ncol = nt * 16 + (lane & 15);  // this lane's N column
        v8f acc[NCH];
        #pragma unroll
        for (int c = 0; c < NCH; ++c)
            acc[c] = (v8f){0.f, 0.f, 0.f, 0.f, 0.f, 0.f, 0.f, 0.f};

        #pragma unroll 4
        for (int k = 0; k < COLS; k += 4) {
            const int kk = k + koff;
            v2f bfrag;                            // B 4x16 fragment of W
            bfrag.x = W[kk * COLS + ncol];
            bfrag.y = W[(kk + 1) * COLS + ncol];
            #pragma unroll
            for (int c = 0; c < NCH; ++c) {       // reuse B across 6 channels
                const v2f afrag = *(const v2f*)&cur[c * ROWS * RSTRIDE +
                                                   mrow * RSTRIDE + kk];
                acc[c] = __builtin_amdgcn_wmma_f32_16x16x4_f32(
                    false, afrag, false, bfrag, (short)0, acc[c], false, false);
            }
        }

        // bias on value channel only, then activation chain rule in-register
        const float bb = B[ncol];
        #pragma unroll
        for (int e = 0; e < 8; ++e) {
            float h, hx, hy, ht, hxx, hyy;
            swish_d2(acc[0][e] + bb, acc[1][e], acc[2][e],
                     acc[3][e], acc[4][e], acc[5][e],
                     h, hx, hy, ht, hxx, hyy);
            const int base = (mbase + e) * RSTRIDE + ncol;
            nxt[0 * ROWS * RSTRIDE + base] = h;
            nxt[1 * ROWS * RSTRIDE + base] = hx;
            nxt[2 * ROWS * RSTRIDE + base] = hy;
            nxt[3 * ROWS * RSTRIDE + base] = ht;
            nxt[4 * ROWS * RSTRIDE + base] = hxx;
            nxt[5 * ROWS * RSTRIDE + base] = hyy;
        }
    }
}

__global__ __launch_bounds__(NTHREADS)
void PINN_Burgers2D_30863634989174_kernel(
        const float* __restrict__ x,  const float* __restrict__ y,
        const float* __restrict__ t,  const float* __restrict__ nu,
        const float* __restrict__ W0, const float* __restrict__ b0,
        const float* __restrict__ W1, const float* __restrict__ b1,
        const float* __restrict__ W2, const float* __restrict__ b2,
        const float* __restrict__ W3, const float* __restrict__ b3,
        const float* __restrict__ W4, const float* __restrict__ b4,
        float* __restrict__ out)
{
    extern __shared__ float smem[];
    float* bufA = smem;
    float* bufB = smem + HBUF;
    float* misc = smem + 2 * HBUF;

    const int tid = threadIdx.x;
    const int g0  = blockIdx.x * ROWS;

    // ---- stage normalized inputs (x,y in [-1,1] already; t,nu rescaled) ----
    if (tid < ROWS) {
        float xv = x[g0 + tid], yv = y[g0 + tid];
        float tv = t[g0 + tid], nv = nu[g0 + tid];
        misc[MISC_IN4 + tid * 4 + 0] = xv;                                 // X
        misc[MISC_IN4 + tid * 4 + 1] = yv;                                 // Y
        misc[MISC_IN4 + tid * 4 + 2] = 2.0f * tv - 1.0f;                   // T
        misc[MISC_IN4 + tid * 4 + 3] = 2.0f * (nv - 0.01f) / 0.09f - 1.0f; // NU
        misc[MISC_NU + tid] = nv;                                          // raw nu
    }
    __syncthreads();

    // ---- layer 0: (4 -> 256), seed AD channels, write bufA ----
    // d/dx seeds: X'=1 ; d/dy: Y'=1 ; d/dt: T'=2 ; nu has no x/y/t seed.
    for (int idx = tid; idx < ROWS * COLS; idx += NTHREADS) {
        int m = idx >> 8, n = idx & 255;
        float w0 = W0[0 * COLS + n], w1 = W0[1 * COLS + n];
        float w2 = W0[2 * COLS + n], w3 = W0[3 * COLS + n];
        float z = misc[MISC_IN4 + m * 4 + 0] * w0 + misc[MISC_IN4 + m * 4 + 1] * w1 +
                  misc[MISC_IN4 + m * 4 + 2] * w2 + misc[MISC_IN4 + m * 4 + 3] * w3 + b0[n];
        float h, hx, hy, ht, hxx, hyy;
        swish_d2(z, w0, w1, 2.0f * w2, 0.0f, 0.0f, h, hx, hy, ht, hxx, hyy);
        int base = m * RSTRIDE + n;
        bufA[0 * ROWS * RSTRIDE + base] = h;
        bufA[1 * ROWS * RSTRIDE + base] = hx;
        bufA[2 * ROWS * RSTRIDE + base] = hy;
        bufA[3 * ROWS * RSTRIDE + base] = ht;
        bufA[4 * ROWS * RSTRIDE + base] = hxx;
        bufA[5 * ROWS * RSTRIDE + base] = hyy;
    }
    __syncthreads();

    // ---- 3 hidden layers, manually instantiated so W1/W2/W3 loads trace
    // straight back to kernel args (global_load path, clean counters) ----
    hidden_layer(bufA, bufB, W1, b1, W2);
    __syncthreads();
    hidden_layer(bufB, bufA, W2, b2, W3);
    __syncthreads();
    hidden_layer(bufA, bufB, W3, b3, nullptr);
    __syncthreads();
    const float* fin = bufB;   // final hidden activations

    // ---- output layer (256 -> 2), all 6 channels; partial dot per k-slice ----
    {
        const int row = tid & 15, slice = tid >> 4;   // 16 slices of 16 k's
        float accv[12];
        #pragma unroll
        for (int j = 0; j < 12; ++j) accv[j] = 0.f;
        for (int k = slice * 16; k < slice * 16 + 16; ++k) {
            const float wu = W4[k * 2 + 0], wv = W4[k * 2 + 1];
            #pragma unroll
            for (int c = 0; c < NCH; ++c) {
                const float hv = fin[c * ROWS * RSTRIDE + row * RSTRIDE + k];
                accv[c * 2 + 0] += hv * wu;
                accv[c * 2 + 1] += hv * wv;
            }
        }
        float* part = misc + MISC_PART;
        #pragma unroll
        for (int j = 0; j < 12; ++j) part[(row * 16 + slice) * 12 + j] = accv[j];
    }
    __syncthreads();

    // ---- reduce + Burgers residual ----
    if (tid < ROWS) {
        const float* part = misc + MISC_PART;
        float v12[12];
        #pragma unroll
        for (int j = 0; j < 12; ++j) {
            float s = 0.f;
            for (int sl = 0; sl < 16; ++sl) s += part[(tid * 16 + sl) * 12 + j];
            v12[j] = s;
        }
        const float u  = v12[0] + b4[0], v  = v12[1] + b4[1];
        const float ux = v12[2],  vx = v12[3];
        const float uy = v12[4],  vy = v12[5];
        const float ut = v12[6],  vt = v12[7];
        const float uxx = v12[8],  vxx = v12[9];
        const float uyy = v12[10], vyy = v12[11];
        const float nv = misc[MISC_NU + tid];
        const float fu = ut + u * ux + v * uy - nv * (uxx + uyy);
        const float fv = vt + u * vx + v * vy - nv * (vxx + vyy);
        const int g = g0 + tid;
        out[2 * g + 0] = fu;
        out[2 * g + 1] = fv;
    }
}

extern "C" void kernel_launch(void* const* d_in, const int* in_sizes, int n_in,
                              void* d_out, int out_size, void* d_ws, size_t ws_size,
                              hipStream_t stream) {
    const float* x  = (const float*)d_in[0];
    const float* y  = (const float*)d_in[1];
    const float* t  = (const float*)d_in[2];
    const float* nu = (const float*)d_in[3];
    const float* W0 = (const float*)d_in[4];
    const float* b0 = (const float*)d_in[5];
    const float* W1 = (const float*)d_in[6];
    const float* b1 = (const float*)d_in[7];
    const float* W2 = (const float*)d_in[8];
    const float* b2 = (const float*)d_in[9];
    const float* W3 = (const float*)d_in[10];
    const float* b3 = (const float*)d_in[11];
    const float* W4 = (const float*)d_in[12];
    const float* b4 = (const float*)d_in[13];
    float* out = (float*)d_out;

    const int N = in_sizes[0];               // 16384 points
    const int blocks = N / ROWS;             // 1024 workgroups of 16 points
    const size_t shmem = (size_t)(2 * HBUF + MISC_FLOATS) * sizeof(float); // ~215 KB

    hipFuncSetAttribute((const void*)PINN_Burgers2D_30863634989174_kernel,
                        hipFuncAttributeMaxDynamicSharedMemorySize, (int)shmem);

    PINN_Burgers2D_30863634989174_kernel<<<blocks, NTHREADS, shmem, stream>>>(
        x, y, t, nu, W0, b0, W1, b1, W2, b2, W3, b3, W4, b4, out);
}